// HR2HK_2448131359490
// MI455X (gfx1250) — compile-verified
//
#include <hip/hip_runtime.h>
#include <math.h>

// Problem constants (from the reference: LI=[0,0,1,1,2] -> NORBS=[1,1,3,3,5])
#define NORB   13
#define NN     200
#define NE     4800
#define NK     8
#define DIM    2600      // NN * NORB
#define PAIRS  169       // NORB * NORB
#define NFEAT  107       // sum_{i<=j} NORBS[i]*NORBS[j]

typedef __attribute__((ext_vector_type(2))) float v2f;
typedef __attribute__((ext_vector_type(8))) float v8f;

// Static gather maps for the 5-block orbital structure.
__constant__ int c_offs[5]  = {0, 1, 2, 5, 8};
__constant__ int c_norbs[5] = {1, 1, 3, 3, 5};
// pair_start[i][j] (row-major blocks over full 13x13 -> 169 edge features)
__constant__ int c_ps[25] = {  0,  1,  2,  5,  8,
                              13, 14, 15, 18, 21,
                              26, 29, 32, 41, 50,
                              65, 68, 71, 80, 89,
                             104,109,114,129,144};
// node_start[i][j] for i<=j (107 node features), lower triangle unused
__constant__ int c_ns[25] = {  0,  1,  2,  5,  8,
                               0, 13, 14, 17, 20,
                               0,  0, 25, 34, 43,
                               0,  0,  0, 58, 67,
                               0,  0,  0,  0, 82};

__device__ __forceinline__ int orb_of(int x) {
    return (x < 1) ? 0 : ((x < 2) ? 1 : ((x < 5) ? 2 : ((x < 8) ? 3 : 4)));
}
__device__ __forceinline__ int pmap(int r, int c) {
    int i = orb_of(r), j = orb_of(c);
    return c_ps[i * 5 + j] + (r - c_offs[i]) * c_norbs[j] + (c - c_offs[j]);
}
__device__ __forceinline__ int nmmap(int r, int c) {
    int i = orb_of(r), j = orb_of(c);
    int ri = r - c_offs[i], cj = c - c_offs[j];
    return (i <= j) ? (c_ns[i * 5 + j] + ri * c_norbs[j] + cj)
                    : (c_ns[j * 5 + i] + cj * c_norbs[i] + ri);
}

// ---------------------------------------------------------------------------
// Kernel 1: zero the full (K,2600,2600) complex array; write 0.5*onsite on the
// node-diagonal blocks. float4 stores (2 complex / thread) -> global_store_b128.
// ---------------------------------------------------------------------------
__global__ void hr2hk_init_diag(const float* __restrict__ nodef, float* __restrict__ H) {
    long long t = (long long)blockIdx.x * blockDim.x + threadIdx.x;
    const long long total = (long long)NK * DIM * (DIM / 2);
    if (t >= total) return;
    int pc  = (int)(t % (DIM / 2));
    long long q = t / (DIM / 2);
    int row = (int)(q % DIM);
    int k   = (int)(q / DIM);
    int col0 = pc * 2;
    int a = row / NORB, r = row % NORB;

    float4 out = make_float4(0.f, 0.f, 0.f, 0.f);
    int b0 = col0 / NORB, c0 = col0 % NORB;
    if (a == b0) out.x = 0.5f * nodef[a * NFEAT + nmmap(r, c0)];
    int col1 = col0 + 1;
    int b1 = col1 / NORB, c1 = col1 % NORB;
    if (a == b1) out.z = 0.5f * nodef[a * NFEAT + nmmap(r, c1)];

    long long base = (((long long)k * DIM + row) * DIM + col0) * 2; // float index
    *(float4*)(H + base) = out;
}

// ---------------------------------------------------------------------------
// Kernel 2: phases[k,e] = exp(-2*pi*i * dot(kpoints[k], ecs[e])) via
// V_WMMA_F32_16X16X4_F32. One wave per 16-edge tile.
//   A (16x4): rows = k-point index (8 valid), K-dim = 3 padded to 4.
//   B (4x16): cols = edges, same half-wave K split.
//   D (16x16 f32): lanes 0-15, acc[v] = row M=v -> all 8 k rows per edge lane.
// Fragment loads are branchless: clamped in-bounds addresses + multiplicative
// zeroing for the padded K row / padded k-point rows.
// Trig: hardware V_SIN/COS_F32 take revolutions (sin(2*pi*x)) -> feed theta
// directly, no OCML range reduction needed (|theta| << 256 here).
// ---------------------------------------------------------------------------
__global__ void hr2hk_phases_wmma(const float* __restrict__ kp,
                                  const float* __restrict__ ecs,
                                  float* __restrict__ ph /* (K,E,2) */) {
    int lane  = threadIdx.x;      // 0..31, wave32
    int m     = lane & 15;
    int kh    = lane >> 4;        // which half of the K(=4) dim this lane holds
    int k0    = kh * 2;           // 0 or 2
    int e     = blockIdx.x * 16 + m;   // NE % 16 == 0 -> always < NE

    // Row mask for the 8 padded k-point rows; column mask for padded K-dim
    // element (K index 3). All loads use clamped, in-bounds addresses.
    float rowm = (m < NK) ? 1.f : 0.f;
    float padm = kh ? 0.f : 1.f;          // second component = K index 1 or 3
    int   ms   = m & (NK - 1);            // clamped k-point row
    int   c1   = kh ? 2 : 1;              // in-bounds second-component column

    v2f a;
    a.x = kp[ms * 3 + k0] * rowm;
    a.y = kp[ms * 3 + c1] * rowm * padm;
    v2f b;
    b.x = ecs[e * 3 + k0];
    b.y = ecs[e * 3 + c1] * padm;

    v8f acc = {0.f, 0.f, 0.f, 0.f, 0.f, 0.f, 0.f, 0.f};
#if __has_builtin(__builtin_amdgcn_wmma_f32_16x16x4_f32)
    // (neg_a, A, neg_b, B, c_mod, C, reuse_a, reuse_b)
    acc = __builtin_amdgcn_wmma_f32_16x16x4_f32(false, a, false, b,
                                                (short)0, acc, false, false);
#else
    if (kh == 0) {
        float b0 = ecs[e * 3 + 0];
        float b1 = ecs[e * 3 + 1];
        float b2 = ecs[e * 3 + 2];
        for (int v = 0; v < NK; ++v)
            acc[v] = kp[v * 3 + 0] * b0 + kp[v * 3 + 1] * b1 + kp[v * 3 + 2] * b2;
    }
#endif

    if (kh == 0) {
#pragma unroll
        for (int v = 0; v < NK; ++v) {
            float theta = acc[v];          // revolutions
#if __has_builtin(__builtin_amdgcn_cosf) && __has_builtin(__builtin_amdgcn_sinf)
            float co = __builtin_amdgcn_cosf(theta);   // v_cos_f32: cos(2*pi*theta)
            float si = __builtin_amdgcn_sinf(theta);   // v_sin_f32: sin(2*pi*theta)
#else
            float ang = 6.28318530717958647692f * theta;
            float co = cosf(ang);
            float si = sinf(ang);
#endif
            long long o = ((long long)v * NE + e) * 2;
            ph[o + 0] = co;    // Re exp(-i*2*pi*theta)
            ph[o + 1] = -si;   // Im exp(-i*2*pi*theta)
        }
    }
}

// ---------------------------------------------------------------------------
// Kernel 3: scatter-add edge contributions. One thread per (k, e, orbital
// pair); duplicates among (src,dst) handled by global_atomic_add_f32.
// ---------------------------------------------------------------------------
__global__ void hr2hk_scatter(const float* __restrict__ ef,
                              const float* __restrict__ ph,
                              const int* __restrict__ eidx,
                              float* __restrict__ H) {
    long long t = (long long)blockIdx.x * blockDim.x + threadIdx.x;
    const long long total = (long long)NK * NE * PAIRS;
    if (t >= total) return;
    int p = (int)(t % PAIRS);
    long long q = t / PAIRS;
    int e = (int)(q % NE);
    int k = (int)(q / NE);
    int r = p / NORB, c = p % NORB;

    int src = eidx[e];
    int dst = eidx[NE + e];
    float hv  = ef[(long long)e * PAIRS + pmap(r, c)];
    long long po = ((long long)k * NE + e) * 2;
    float pre = ph[po + 0];
    float pim = ph[po + 1];

    long long row = (long long)src * NORB + r;
    long long col = (long long)dst * NORB + c;
    long long base = (((long long)k * DIM + row) * DIM + col) * 2;
    atomicAdd(H + base,     pre * hv);
    atomicAdd(H + base + 1, pim * hv);
}

// ---------------------------------------------------------------------------
// Kernel 4: in-place Hermitianize H = H + conj(H^T) per k. Upper-triangular
// 16x16 tile grid; both the (i,j) and (j,i) tiles are staged in LDS so every
// global read and write stays coalesced. Each element read+written once.
// ---------------------------------------------------------------------------
__global__ void hr2hk_hermitianize(float* __restrict__ H) {
    int ti = blockIdx.x, tj = blockIdx.y, k = blockIdx.z;
    if (tj < ti) return;
    __shared__ float2 As[16][17];
    __shared__ float2 Bs[16][17];
    int tx = threadIdx.x, ty = threadIdx.y;
    int R0 = ti * 16, C0 = tj * 16;
    float2* Hc = (float2*)H;
    long long kbase = (long long)k * DIM * DIM;
    float2 z = make_float2(0.f, 0.f);

    int ra = R0 + ty, ca = C0 + tx;
    As[ty][tx] = (ra < DIM && ca < DIM) ? Hc[kbase + (long long)ra * DIM + ca] : z;
    int rb = C0 + ty, cb = R0 + tx;
    Bs[ty][tx] = (rb < DIM && cb < DIM) ? Hc[kbase + (long long)rb * DIM + cb] : z;
    __syncthreads();

    if (ra < DIM && ca < DIM) {
        float2 tv = As[ty][tx], u = Bs[tx][ty];
        Hc[kbase + (long long)ra * DIM + ca] = make_float2(tv.x + u.x, tv.y - u.y);
    }
    if (ti != tj && rb < DIM && cb < DIM) {
        float2 tv = Bs[ty][tx], u = As[tx][ty];
        Hc[kbase + (long long)rb * DIM + cb] = make_float2(tv.x + u.x, tv.y - u.y);
    }
}

extern "C" void kernel_launch(void* const* d_in, const int* in_sizes, int n_in,
                              void* d_out, int out_size, void* d_ws, size_t ws_size,
                              hipStream_t stream) {
    (void)in_sizes; (void)n_in; (void)out_size; (void)ws_size;
    const float* edge_features   = (const float*)d_in[0];  // (E, 169)
    const float* node_features   = (const float*)d_in[1];  // (N, 107)
    const float* kpoints         = (const float*)d_in[2];  // (K, 3)
    const float* edge_cell_shift = (const float*)d_in[3];  // (E, 3)
    const int*   edge_index      = (const int*)d_in[4];    // (2, E)
    // d_in[5] = atom_types, unused by the reference math.

    float* H  = (float*)d_out;  // (K, 2600, 2600) complex64 as interleaved f32
    float* ph = (float*)d_ws;   // (K, E) complex phases: 307,200 bytes

    {   // zero + diagonal blocks (full 432 MB write, b128 stores)
        long long total = (long long)NK * DIM * (DIM / 2);
        int blk = 256;
        unsigned grid = (unsigned)((total + blk - 1) / blk);
        hr2hk_init_diag<<<grid, blk, 0, stream>>>(node_features, H);
    }
    {   // phases via WMMA f32 16x16x4: 300 waves
        hr2hk_phases_wmma<<<dim3(NE / 16), dim3(32), 0, stream>>>(
            kpoints, edge_cell_shift, ph);
    }
    {   // edge scatter-add
        long long total = (long long)NK * NE * PAIRS;
        int blk = 256;
        unsigned grid = (unsigned)((total + blk - 1) / blk);
        hr2hk_scatter<<<grid, blk, 0, stream>>>(edge_features, ph, edge_index, H);
    }
    {   // Hermitianize in place
        int nt = (DIM + 15) / 16;  // 163
        hr2hk_hermitianize<<<dim3(nt, nt, NK), dim3(16, 16), 0, stream>>>(H);
    }
}